// VectorQuantizer_11879879544243
// MI455X (gfx1250) — compile-verified
//
#include <hip/hip_runtime.h>

#define NEMB 1024
#define CDIM 64
#define HW   4096            // 64*64
#define NROWS 65536          // 16 * HW
#define OUTELEMS 4194304     // NROWS * CDIM

typedef __attribute__((ext_vector_type(2))) float v2f;
typedef __attribute__((ext_vector_type(8))) float v8f;

// ||e_k||^2 for all 1024 codes
__global__ __launch_bounds__(256) void vq_norms(const float* __restrict__ E,
                                                float* __restrict__ eNorm) {
    int k = blockIdx.x * blockDim.x + threadIdx.x;
    if (k < NEMB) {
        const float* row = E + k * CDIM;
        float s = 0.f;
#pragma unroll
        for (int c = 0; c < CDIM; ++c) { float v = row[c]; s += v * v; }
        eNorm[k] = s;
    }
}

// Argmin over codes via chained V_WMMA_F32_16X16X4_F32.
// Block = 128 threads (4 waves); each wave owns two 16-row tiles of flat_z.
// Codebook staged through LDS in 4 stages of 256 codes (64 KB each),
// pair-interleaved transposed: lds2[cpair*512 + k*2 + (c&1)] = E[k][c].
__global__ __launch_bounds__(128) void vq_argmin(const float* __restrict__ z,
                                                 const float* __restrict__ E,
                                                 const float* __restrict__ eNorm,
                                                 int* __restrict__ outIdx) {
    __shared__ float lds2[32 * 512];   // 16384 floats = 64 KB

    const int tid  = threadIdx.x;
    const int l    = tid & 31;
    const int hi   = l >> 4;           // which 16-lane half
    const int lo16 = l & 15;
    const int wave = tid >> 5;
    const int t0   = blockIdx.x * 128 + wave * 32;  // first row tile base
    const int t1   = t0 + 16;                       // second row tile base

    // ---- Preload WMMA-A operands (flat_z rows, fp32 16x4 layout) ----
    // A layout: lanes 0-15 hold M=lane; VGPR0 = K in {0 (lo half), 2 (hi half)},
    // VGPR1 = K+1.  flat_z[n][c] = z[(n>>12)*C*HW + c*HW + (n&4095)].
    v2f a0[16], a1[16];
    {
        int n0 = t0 + lo16;
        int n1 = t1 + lo16;
        const float* zp0 = z + (n0 >> 12) * (CDIM * HW) + (n0 & (HW - 1));
        const float* zp1 = z + (n1 >> 12) * (CDIM * HW) + (n1 & (HW - 1));
#pragma unroll
        for (int s = 0; s < 16; ++s) {
            int cx = 4 * s + 2 * hi;
            a0[s].x = zp0[cx * HW];
            a0[s].y = zp0[(cx + 1) * HW];
            a1[s].x = zp1[cx * HW];
            a1[s].y = zp1[(cx + 1) * HW];
        }
    }

    float best0[8], best1[8];
    int   bidx0[8], bidx1[8];
#pragma unroll
    for (int j = 0; j < 8; ++j) {
        best0[j] = 3.4e38f; best1[j] = 3.4e38f;
        bidx0[j] = 0;       bidx1[j] = 0;
    }

    for (int stage = 0; stage < 4; ++stage) {
        const int stageBase = stage * 256;
        __syncthreads();                       // previous stage fully consumed
        for (int i = tid; i < 256 * CDIM; i += 128) {
            int k = i >> 6;
            int c = i & 63;
            lds2[(c >> 1) * 512 + k * 2 + (c & 1)] = E[(stageBase + k) * CDIM + c];
        }
        __syncthreads();

        for (int chunk = 0; chunk < 16; ++chunk) {
            const int kLoc  = chunk * 16 + lo16;     // this lane's code (local)
            const int kGlob = stageBase + kLoc;      // this lane's code (global)
            const float en  = eNorm[kGlob];

            v8f acc0 = {0.f,0.f,0.f,0.f,0.f,0.f,0.f,0.f};
            v8f acc1 = {0.f,0.f,0.f,0.f,0.f,0.f,0.f,0.f};
#pragma unroll
            for (int s = 0; s < 16; ++s) {
                // B operand (4x16 fp32): lane = N (code), VGPR0/1 = K pair.
                v2f bb = *(const v2f*)&lds2[(2 * s + hi) * 512 + kLoc * 2];
                acc0 = __builtin_amdgcn_wmma_f32_16x16x4_f32(
                    false, a0[s], false, bb, (short)0, acc0, false, false);
                acc1 = __builtin_amdgcn_wmma_f32_16x16x4_f32(
                    false, a1[s], false, bb, (short)0, acc1, false, false);
            }
            // dist = ||e||^2 - 2*score  (||z||^2 constant per row -> drop)
#pragma unroll
            for (int j = 0; j < 8; ++j) {
                float d0 = en - 2.f * acc0[j];
                if (d0 < best0[j]) { best0[j] = d0; bidx0[j] = kGlob; }
                float d1 = en - 2.f * acc1[j];
                if (d1 < best1[j]) { best1[j] = d1; bidx1[j] = kGlob; }
            }
        }
    }

    // ---- Cross-lane (min, argmin) reduce within each 16-lane half ----
#pragma unroll
    for (int mask = 1; mask <= 8; mask <<= 1) {
#pragma unroll
        for (int j = 0; j < 8; ++j) {
            float od = __shfl_xor(best0[j], mask, 32);
            int   oi = __shfl_xor(bidx0[j], mask, 32);
            if (od < best0[j] || (od == best0[j] && oi < bidx0[j])) {
                best0[j] = od; bidx0[j] = oi;
            }
            od = __shfl_xor(best1[j], mask, 32);
            oi = __shfl_xor(bidx1[j], mask, 32);
            if (od < best1[j] || (od == best1[j] && oi < bidx1[j])) {
                best1[j] = od; bidx1[j] = oi;
            }
        }
    }

    // C/D layout: VGPR j holds row M = j + 8*hi
    if (lo16 == 0) {
#pragma unroll
        for (int j = 0; j < 8; ++j) {
            outIdx[t0 + j + 8 * hi] = bidx0[j];
            outIdx[t1 + j + 8 * hi] = bidx1[j];
        }
    }
}

// Gather e[idx] back into BCHW layout, write straight-through value,
// accumulate per-block sum of (q - z)^2 (deterministic partials).
__global__ __launch_bounds__(256) void vq_gather(const float* __restrict__ z,
                                                 const float* __restrict__ E,
                                                 const int* __restrict__ idxArr,
                                                 float* __restrict__ out,
                                                 float* __restrict__ partial) {
    __shared__ float red[256];
    int e  = blockIdx.x * 256 + threadIdx.x;   // element index in BCHW
    int b  = e >> 18;                          // 262144 elems per batch image
    int r  = e & 262143;
    int c  = r >> 12;
    int hw = r & 4095;
    int n  = (b << 12) | hw;                   // flat_z row index
    int idx = idxArr[n];
    float q    = E[idx * CDIM + c];
    float zv   = z[e];
    float diff = q - zv;                       // quantized - z
    out[e] = zv + diff;                        // z + stopgrad(quantized - z)
    red[threadIdx.x] = diff * diff;
    __syncthreads();
    for (int sft = 128; sft > 0; sft >>= 1) {
        if (threadIdx.x < sft) red[threadIdx.x] += red[threadIdx.x + sft];
        __syncthreads();
    }
    if (threadIdx.x == 0) partial[blockIdx.x] = red[0];
}

// loss = q_latent + 0.25*e_latent = 1.25 * mean((q - z)^2)
__global__ __launch_bounds__(256) void vq_loss(const float* __restrict__ partial,
                                               float* __restrict__ out) {
    __shared__ float red[256];
    float s = 0.f;
    for (int i = threadIdx.x; i < 16384; i += 256) s += partial[i];
    red[threadIdx.x] = s;
    __syncthreads();
    for (int sft = 128; sft > 0; sft >>= 1) {
        if (threadIdx.x < sft) red[threadIdx.x] += red[threadIdx.x + sft];
        __syncthreads();
    }
    if (threadIdx.x == 0)
        out[OUTELEMS] = 1.25f * red[0] / (float)OUTELEMS;
}

extern "C" void kernel_launch(void* const* d_in, const int* in_sizes, int n_in,
                              void* d_out, int out_size, void* d_ws, size_t ws_size,
                              hipStream_t stream) {
    const float* z = (const float*)d_in[0];   // [16, 64, 64, 64] fp32
    const float* E = (const float*)d_in[1];   // [1024, 64] fp32
    float* out = (float*)d_out;               // 4194304 quantized_st + 1 loss

    char* ws = (char*)d_ws;
    int*   wsIdx     = (int*)ws;                                   // 65536 ints
    float* wsPartial = (float*)(ws + NROWS * sizeof(int));         // 16384 floats
    float* wsNorm    = (float*)(ws + NROWS * sizeof(int)
                                   + 16384 * sizeof(float));       // 1024 floats

    vq_norms <<<4,     256, 0, stream>>>(E, wsNorm);
    vq_argmin<<<512,   128, 0, stream>>>(z, E, wsNorm, wsIdx);
    vq_gather<<<16384, 256, 0, stream>>>(z, E, wsIdx, out, wsPartial);
    vq_loss  <<<1,     256, 0, stream>>>(wsPartial, out);
}